// GDE_81758997447375
// MI455X (gfx1250) — compile-verified
//
#include <hip/hip_runtime.h>
#include <math.h>

#define GDE_N 50000
#define GDE_E 800000

typedef float v2f __attribute__((ext_vector_type(2)));
typedef float v8f __attribute__((ext_vector_type(8)));

// ---------------------------------------------------------------------------
// Dense node matmul  Y[n,Ncols] = X[n,K] @ G[K,Ncols]  via V_WMMA_F32_16X16X4_F32
// One wave32 per 16x16 output tile. EXEC is all-ones (uniform control flow).
// A layout (ISA 7.12.2, 32-bit A 16x4): lane m=lane&15 holds row m;
//   half=lane>>4 selects K pair {2h, 2h+1} in the two A VGPRs.
// B/C/D: row striped across lanes within a VGPR; D VGPR r -> row r + 8*half.
// ---------------------------------------------------------------------------
__global__ __launch_bounds__(32) void gde_mm_wmma(const float* __restrict__ X,
                                                  const float* __restrict__ G,
                                                  float* __restrict__ Y,
                                                  int K, int Ncols) {
  const int lane = threadIdx.x;
  const int m    = lane & 15;
  const int half = lane >> 4;
  const int row  = blockIdx.x * 16 + m;           // A row for this lane
  const int col  = blockIdx.y * 16 + m;           // B/D column for this lane
  const float* __restrict__ xr = X + (size_t)row * K;
  v8f acc = {0.f, 0.f, 0.f, 0.f, 0.f, 0.f, 0.f, 0.f};
  for (int k = 0; k < K; k += 4) {
    const int ka = k + 2 * half;
    v2f a; a.x = xr[ka];
           a.y = xr[ka + 1];
    v2f b; b.x = G[(size_t)ka       * Ncols + col];
           b.y = G[(size_t)(ka + 1) * Ncols + col];
    acc = __builtin_amdgcn_wmma_f32_16x16x4_f32(false, a, false, b,
                                                (short)0, acc, false, false);
  }
  const int rbase = blockIdx.x * 16 + half * 8;
#pragma unroll
  for (int r = 0; r < 8; ++r)
    Y[(size_t)(rbase + r) * Ncols + col] = acc[r];
}

// Build G = [W0 | W1-W0 | R] : [Cin, 3*Cout]
__global__ void gde_prep_g(float* __restrict__ G, const float* __restrict__ Wp,
                           const float* __restrict__ R, int Cin, int Cout) {
  int idx = blockIdx.x * blockDim.x + threadIdx.x;
  int Ncols = 3 * Cout;
  int tot = Cin * Ncols;
  if (idx >= tot) return;
  int r = idx / Ncols, c = idx % Ncols;
  float v;
  if (c < Cout)            v = Wp[r * Cout + c];
  else if (c < 2 * Cout)   v = Wp[(size_t)Cin * Cout + r * Cout + (c - Cout)]
                             - Wp[r * Cout + (c - Cout)];
  else                     v = R[r * Cout + (c - 2 * Cout)];
  G[idx] = v;
}

__global__ void gde_deg(float* __restrict__ deg, const int* __restrict__ dst, int E) {
  int e = blockIdx.x * blockDim.x + threadIdx.x;
  if (e < E) atomicAdd(&deg[dst[e]], 1.0f);
}

__global__ void gde_invdeg(float* __restrict__ d, int n) {
  int i = blockIdx.x * blockDim.x + threadIdx.x;
  if (i < n) d[i] = 1.0f / fmaxf(d[i], 1.0f);
}

// Edge scatter: agg[dst] += Z0[src] + p * Z1[src]   (4 channels per thread)
__global__ void gde_edge_agg(float* __restrict__ agg, const float* __restrict__ Z,
                             const int* __restrict__ src, const int* __restrict__ dst,
                             const float* __restrict__ p, int Cout, long long total) {
  long long t = (long long)blockIdx.x * blockDim.x + threadIdx.x;
  if (t >= total) return;
  const int groups = Cout >> 2;
  const int e = (int)(t / groups);
  const int c = (int)(t % groups) * 4;
  const int s = src[e], d = dst[e];
  const float pe = p[e];
  const int ldz = 3 * Cout;
  const float4 z0 = *(const float4*)(Z + (size_t)s * ldz + c);
  const float4 z1 = *(const float4*)(Z + (size_t)s * ldz + Cout + c);
  float* ap = agg + (size_t)d * Cout + c;
  atomicAdd(ap + 0, z0.x + pe * z1.x);
  atomicAdd(ap + 1, z0.y + pe * z1.y);
  atomicAdd(ap + 2, z0.z + pe * z1.z);
  atomicAdd(ap + 3, z0.w + pe * z1.w);
}

// out = agg * invdeg + ZR + b  (+ optional tanh)
__global__ void gde_finalize(float* __restrict__ out, const float* __restrict__ agg,
                             const float* __restrict__ Z, const float* __restrict__ bias,
                             const float* __restrict__ inv, int Cout, int act,
                             long long total) {
  long long idx = (long long)blockIdx.x * blockDim.x + threadIdx.x;
  if (idx >= total) return;
  int i = (int)(idx / Cout), c = (int)(idx % Cout);
  float v = agg[idx] * inv[i] + Z[(size_t)i * 3 * Cout + 2 * Cout + c] + bias[c];
  if (act) v = tanhf(v);
  out[idx] = v;
}

// ACC = (set? w*K : ACC + w*K);  YT = H + c*K
__global__ void gde_rk_stage(float* __restrict__ ACC, float* __restrict__ YT,
                             const float* __restrict__ H, const float* __restrict__ K,
                             float w, float c, int set_acc, long long n) {
  long long i = (long long)blockIdx.x * blockDim.x + threadIdx.x;
  if (i >= n) return;
  float k = K[i];
  ACC[i] = set_acc ? (w * k) : (ACC[i] + w * k);
  YT[i] = H[i] + c * k;
}

// H = H + 0.5 * (ACC + K)      (T_END/6 = 0.5)
__global__ void gde_rk_final(float* __restrict__ H, const float* __restrict__ ACC,
                             const float* __restrict__ K, long long n) {
  long long i = (long long)blockIdx.x * blockDim.x + threadIdx.x;
  if (i >= n) return;
  H[i] = H[i] + 0.5f * (ACC[i] + K[i]);
}

__global__ void gde_log_softmax16(float* __restrict__ out, int n) {
  int i = blockIdx.x * blockDim.x + threadIdx.x;
  if (i >= n) return;
  float v[16];
  float mx = -3.402823466e+38f;
  float* row = out + (size_t)i * 16;
#pragma unroll
  for (int c = 0; c < 16; ++c) { v[c] = row[c]; mx = fmaxf(mx, v[c]); }
  float s = 0.f;
#pragma unroll
  for (int c = 0; c < 16; ++c) s += __expf(v[c] - mx);
  float ls = __logf(s);
#pragma unroll
  for (int c = 0; c < 16; ++c) row[c] = v[c] - mx - ls;
}

extern "C" void kernel_launch(void* const* d_in, const int* in_sizes, int n_in,
                              void* d_out, int out_size, void* d_ws, size_t ws_size,
                              hipStream_t stream) {
  const float* x   = (const float*)d_in[0];
  const float* p   = (const float*)d_in[1];
  const int*   src = (const int*)d_in[2];
  const int*   dst = (const int*)d_in[3];
  const float* W1 = (const float*)d_in[4],  *R1 = (const float*)d_in[5],  *b1 = (const float*)d_in[6];
  const float* Wa = (const float*)d_in[7],  *Ra = (const float*)d_in[8],  *ba = (const float*)d_in[9];
  const float* Wb = (const float*)d_in[10], *Rb = (const float*)d_in[11], *bb = (const float*)d_in[12];
  const float* W2 = (const float*)d_in[13], *R2 = (const float*)d_in[14], *b2 = (const float*)d_in[15];
  float* out = (float*)d_out;

  // ---- carve workspace (floats) ----
  float* W = (float*)d_ws;
  size_t o = 0;
  float* Z   = W + o; o += (size_t)GDE_N * 192;   // fused [Z0|Z1|ZR], max 3*64 cols
  float* AGG = W + o; o += (size_t)GDE_N * 64;
  float* H   = W + o; o += (size_t)GDE_N * 64;
  float* YT  = W + o; o += (size_t)GDE_N * 64;
  float* U   = W + o; o += (size_t)GDE_N * 64;
  float* Kb  = W + o; o += (size_t)GDE_N * 64;
  float* ACC = W + o; o += (size_t)GDE_N * 64;
  float* INV = W + o; o += (size_t)GDE_N;
  float* GC1 = W + o; o += (size_t)256 * 192;
  float* GCA = W + o; o += (size_t)64 * 192;
  float* GCB = W + o; o += (size_t)64 * 192;
  float* GC2 = W + o; o += (size_t)64 * 48;
  (void)ws_size; (void)n_in; (void)in_sizes; (void)out_size;

  // ---- prep fused weights G = [W0 | W1-W0 | R] ----
  gde_prep_g<<<(256 * 192 + 255) / 256, 256, 0, stream>>>(GC1, W1, R1, 256, 64);
  gde_prep_g<<<(64 * 192 + 255) / 256, 256, 0, stream>>>(GCA, Wa, Ra, 64, 64);
  gde_prep_g<<<(64 * 192 + 255) / 256, 256, 0, stream>>>(GCB, Wb, Rb, 64, 64);
  gde_prep_g<<<(64 * 48 + 255) / 256, 256, 0, stream>>>(GC2, W2, R2, 64, 16);

  // ---- degree -> 1/max(deg,1) ----
  hipMemsetAsync(INV, 0, sizeof(float) * GDE_N, stream);
  gde_deg<<<(GDE_E + 255) / 256, 256, 0, stream>>>(INV, dst, GDE_E);
  gde_invdeg<<<(GDE_N + 255) / 256, 256, 0, stream>>>(INV, GDE_N);

  // ---- one SplineConv: Z = y@G ; agg = scatter(Z0 + p*Z1); out = agg/deg + ZR + b ----
  auto spline = [&](const float* y, int Cin, int Cout, const float* G,
                    const float* bias, float* ob, int act) {
    const int N3 = 3 * Cout;
    dim3 grid(GDE_N / 16, N3 / 16);
    gde_mm_wmma<<<grid, 32, 0, stream>>>(y, G, Z, Cin, N3);
    hipMemsetAsync(AGG, 0, sizeof(float) * (size_t)GDE_N * Cout, stream);
    long long tot = (long long)GDE_E * (Cout / 4);
    gde_edge_agg<<<(int)((tot + 255) / 256), 256, 0, stream>>>(AGG, Z, src, dst, p, Cout, tot);
    long long tn = (long long)GDE_N * Cout;
    gde_finalize<<<(int)((tn + 255) / 256), 256, 0, stream>>>(ob, AGG, Z, bias, INV, Cout, act, tn);
  };
  auto odefunc = [&](const float* y, float* k) {  // f(y) = spline_b(spline_a(y))
    spline(y, 64, 64, GCA, ba, U, 0);
    spline(U, 64, 64, GCB, bb, k, 0);
  };

  const long long tn = (long long)GDE_N * 64;
  const int eb = (int)((tn + 255) / 256);

  // h = tanh(conv1(x))
  spline(x, 256, 64, GC1, b1, H, 1);

  // single RK4 step over [0, 3]: T=3, T/2=1.5, T/6=0.5
  odefunc(H, Kb);                                                        // k1
  gde_rk_stage<<<eb, 256, 0, stream>>>(ACC, YT, H, Kb, 1.0f, 1.5f, 1, tn);
  odefunc(YT, Kb);                                                       // k2
  gde_rk_stage<<<eb, 256, 0, stream>>>(ACC, YT, H, Kb, 2.0f, 1.5f, 0, tn);
  odefunc(YT, Kb);                                                       // k3
  gde_rk_stage<<<eb, 256, 0, stream>>>(ACC, YT, H, Kb, 2.0f, 3.0f, 0, tn);
  odefunc(YT, Kb);                                                       // k4
  gde_rk_final<<<eb, 256, 0, stream>>>(H, ACC, Kb, tn);

  // out = log_softmax(tanh(conv2(h)))
  spline(H, 64, 16, GC2, b2, out, 1);
  gde_log_softmax16<<<(GDE_N + 255) / 256, 256, 0, stream>>>(out, GDE_N);
}